// RGCNLinkPredictor_13451837571727
// MI455X (gfx1250) — compile-verified
//
#include <hip/hip_runtime.h>
#include <hip/hip_bf16.h>

typedef __attribute__((ext_vector_type(2))) float v2f;
typedef __attribute__((ext_vector_type(8))) float v8f;

#define N_NODES 60000
#define N_HERB 5000
#define DIM 128
#define N_REL 6
#define N_BASES 30
#define N_EDGES 600000
#define N_PAIRS 40000

// ---------------- zero workspace ----------------
__global__ void k_zero(float* __restrict__ p, size_t n) {
    size_t i = (size_t)blockIdx.x * blockDim.x + threadIdx.x;
    size_t stride = (size_t)gridDim.x * blockDim.x;
    for (; i < n; i += stride) p[i] = 0.f;
}

// ---------------- W_r = sum_b comp[r,b] * basis[b]  ([R,D,D]) ----------------
__global__ void k_relw(float* __restrict__ W, const float* __restrict__ basis,
                       const float* __restrict__ comp) {
    int i = blockIdx.x * blockDim.x + threadIdx.x;
    if (i >= N_REL * DIM * DIM) return;
    int r = i >> 14;           // /16384
    int idx = i & 16383;
    float acc = 0.f;
    #pragma unroll 5
    for (int b = 0; b < N_BASES; ++b)
        acc += comp[r * N_BASES + b] * basis[(size_t)b * DIM * DIM + idx];
    W[i] = acc;
}

// ---------------- edge scatter: agg[(dst*R+rel)*D + d] += x[src*D + d] ----------------
__global__ void k_scatter(float* __restrict__ agg, float* __restrict__ cnt,
                          const float* __restrict__ x,
                          const int* __restrict__ src, const int* __restrict__ dst,
                          const int* __restrict__ etype, int nedges, int do_cnt) {
    long long t = (long long)blockIdx.x * blockDim.x + threadIdx.x;
    int e = (int)(t >> 5);
    int lane = (int)(t & 31);
    if (e >= nedges) return;
    int s = src[e], d = dst[e], r = etype[e];
    const float* xs = x + (size_t)s * DIM;
    float* ag = agg + ((size_t)d * N_REL + r) * DIM;
    #pragma unroll
    for (int k = 0; k < 4; ++k) {
        int c = lane + k * 32;                 // coalesced per wave
        atomicAdd(&ag[c], xs[c]);
    }
    if (do_cnt && lane == 0) atomicAdd(&cnt[(size_t)d * N_REL + r], 1.0f);
}

// ---------------- RGCN node transform via V_WMMA_F32_16X16X4_F32 ----------------
// out[i,:] = sum_{r} mean[i,r,:] @ W_r + x[i,:] @ root + bias   (optional ReLU)
#define ASTRIDE 132   // 128 + 4 pad -> bank = (4*m + k) % 64, conflict-free A reads
__global__ __launch_bounds__(256) void k_rgcn(
    float* __restrict__ out, const float* __restrict__ agg, const float* __restrict__ cnt,
    const float* __restrict__ x, const float* __restrict__ W,
    const float* __restrict__ root, const float* __restrict__ bias, int relu) {
    __shared__ float As[N_REL + 1][16][ASTRIDE];
    __shared__ float icnt[16][N_REL];
    const int t = threadIdx.x;
    const int base = blockIdx.x * 16;

    if (t < 16 * N_REL) {
        int m = t / N_REL, r = t % N_REL;
        icnt[m][r] = 1.0f / fmaxf(cnt[(size_t)(base + m) * N_REL + r], 1.0f);
    }
    __syncthreads();

    // stage A tiles: 6 relation means + root input, 16 x 128 each
    for (int idx = t; idx < (N_REL + 1) * 16 * DIM; idx += 256) {
        int s = idx >> 11;            // /2048
        int rem = idx & 2047;
        int m = rem >> 7, d = rem & 127;
        float v;
        if (s < N_REL)
            v = agg[((size_t)(base + m) * N_REL + s) * DIM + d] * icnt[m][s];
        else
            v = x[(size_t)(base + m) * DIM + d];
        As[s][m][d] = v;
    }
    __syncthreads();

    const int wave = t >> 5, lane = t & 31;
    const int nb = wave * 16;              // 8 waves -> 8 column tiles of 16
    const int m = lane & 15;
    const int khalf = (lane >> 4) << 1;    // lanes 16-31 hold K+2,K+3
    const int n = nb + m;

    v8f c = {};
    for (int s = 0; s <= N_REL; ++s) {
        const float* B = (s < N_REL) ? (W + (size_t)s * DIM * DIM) : root;
        #pragma unroll 4
        for (int k0 = 0; k0 < DIM; k0 += 4) {
            int kb = k0 + khalf;
            v2f a; a.x = As[s][m][kb];            a.y = As[s][m][kb + 1];
            v2f b; b.x = B[(size_t)kb * DIM + n]; b.y = B[(size_t)(kb + 1) * DIM + n];
            c = __builtin_amdgcn_wmma_f32_16x16x4_f32(false, a, false, b,
                                                      (short)0, c, false, false);
        }
    }
    const float bv = bias[n];
    #pragma unroll
    for (int v = 0; v < 8; ++v) {
        int M = v + ((lane >> 4) << 3);
        float o = c[v] + bv;
        if (relu) o = fmaxf(o, 0.f);
        out[(size_t)(base + M) * DIM + n] = o;
    }
}

// ---------------- pair scorer: feat=[h,p,h*p,|h-p|] -> 512x128 WMMA -> ReLU -> dot ----------------
#define FSTRIDE 516   // 512 + 4 pad
#define HSTRIDE 132
__global__ __launch_bounds__(256) void k_score(
    float* __restrict__ out, const float* __restrict__ x2,
    const int* __restrict__ h_idx, const int* __restrict__ p_idx,
    const float* __restrict__ sw1, const float* __restrict__ sb1,
    const float* __restrict__ sw2, const float* __restrict__ sb2) {
    __shared__ float F[16][FSTRIDE];
    __shared__ float H[16][HSTRIDE];
    const int t = threadIdx.x;
    const int pb = blockIdx.x * 16;

    for (int idx = t; idx < 16 * DIM; idx += 256) {
        int m = idx >> 7, d = idx & 127;
        int hi = h_idx[pb + m], pi = p_idx[pb + m];
        float h = x2[(size_t)hi * DIM + d];
        float p = x2[(size_t)(N_HERB + pi) * DIM + d];
        F[m][d]          = h;
        F[m][DIM + d]    = p;
        F[m][2*DIM + d]  = h * p;
        F[m][3*DIM + d]  = fabsf(h - p);
    }
    __syncthreads();

    const int wave = t >> 5, lane = t & 31;
    const int nb = wave * 16;
    const int m = lane & 15;
    const int khalf = (lane >> 4) << 1;
    const int n = nb + m;

    v8f c = {};
    #pragma unroll 4
    for (int k0 = 0; k0 < 4 * DIM; k0 += 4) {
        int kb = k0 + khalf;
        v2f a; a.x = F[m][kb];                  a.y = F[m][kb + 1];
        v2f b; b.x = sw1[(size_t)kb * DIM + n]; b.y = sw1[(size_t)(kb + 1) * DIM + n];
        c = __builtin_amdgcn_wmma_f32_16x16x4_f32(false, a, false, b,
                                                  (short)0, c, false, false);
    }
    const float bv = sb1[n];
    #pragma unroll
    for (int v = 0; v < 8; ++v) {
        int M = v + ((lane >> 4) << 3);
        H[M][n] = fmaxf(c[v] + bv, 0.f);
    }
    __syncthreads();

    if (t < 16) {
        float acc = sb2[0];
        for (int j = 0; j < DIM; ++j) acc += H[t][j] * sw2[j];
        out[pb + t] = acc;
    }
}

extern "C" void kernel_launch(void* const* d_in, const int* in_sizes, int n_in,
                              void* d_out, int out_size, void* d_ws, size_t ws_size,
                              hipStream_t stream) {
    const int*   edge_index = (const int*)d_in[0];
    const int*   edge_type  = (const int*)d_in[1];
    const int*   h_idx      = (const int*)d_in[2];
    const int*   p_idx      = (const int*)d_in[3];
    const float* node_emb   = (const float*)d_in[4];
    const float* basis1     = (const float*)d_in[5];
    const float* comp1      = (const float*)d_in[6];
    const float* root1      = (const float*)d_in[7];
    const float* bias1      = (const float*)d_in[8];
    const float* basis2     = (const float*)d_in[9];
    const float* comp2      = (const float*)d_in[10];
    const float* root2      = (const float*)d_in[11];
    const float* bias2      = (const float*)d_in[12];
    const float* sw1        = (const float*)d_in[13];
    const float* sb1        = (const float*)d_in[14];
    const float* sw2        = (const float*)d_in[15];
    const float* sb2        = (const float*)d_in[16];

    const int* src = edge_index;              // edge_index[0,:]
    const int* dst = edge_index + N_EDGES;    // edge_index[1,:]

    float* ws = (float*)d_ws;
    size_t off = 0;
    float* agg = ws + off; off += (size_t)N_NODES * N_REL * DIM; // 46.08M f
    float* cnt = ws + off; off += (size_t)N_NODES * N_REL;       // contiguous with agg
    float* W   = ws + off; off += (size_t)N_REL * DIM * DIM;
    float* x1  = ws + off; off += (size_t)N_NODES * DIM;
    float* x2  = ws + off; off += (size_t)N_NODES * DIM;

    const long long scatter_threads = (long long)N_EDGES * 32;
    const unsigned scatter_blocks = (unsigned)((scatter_threads + 255) / 256);
    const unsigned relw_blocks = (N_REL * DIM * DIM + 255) / 256;

    // ---- layer 1 ----
    k_zero<<<2048, 256, 0, stream>>>(agg, (size_t)N_NODES * N_REL * DIM + (size_t)N_NODES * N_REL);
    k_relw<<<relw_blocks, 256, 0, stream>>>(W, basis1, comp1);
    k_scatter<<<scatter_blocks, 256, 0, stream>>>(agg, cnt, node_emb, src, dst, edge_type, N_EDGES, 1);
    k_rgcn<<<N_NODES / 16, 256, 0, stream>>>(x1, agg, cnt, node_emb, W, root1, bias1, 1);

    // ---- layer 2 (counts unchanged, reuse) ----
    k_zero<<<2048, 256, 0, stream>>>(agg, (size_t)N_NODES * N_REL * DIM);
    k_relw<<<relw_blocks, 256, 0, stream>>>(W, basis2, comp2);
    k_scatter<<<scatter_blocks, 256, 0, stream>>>(agg, cnt, x1, src, dst, edge_type, N_EDGES, 0);
    k_rgcn<<<N_NODES / 16, 256, 0, stream>>>(x2, agg, cnt, x1, W, root2, bias2, 0);

    // ---- scorer ----
    k_score<<<N_PAIRS / 16, 256, 0, stream>>>((float*)d_out, x2, h_idx, p_idx, sw1, sb1, sw2, sb2);
}